// MutualInformation2D_38654705664143
// MI455X (gfx1250) — compile-verified
//
#include <hip/hip_runtime.h>
#include <math.h>

#define BINS    32
#define NBINS2  (BINS * BINS)            // 1024
#define BATCH   32
#define N_PER   (2 * 512 * 512)          // 524288 elements per sample
#define F4_PER  (N_PER / 4)              // 131072 float4 per sample
#define CHUNKS  64                       // blocks per sample
#define HTHREADS 256
#define F4_PER_BLOCK  (F4_PER / CHUNKS)        // 2048
#define F4_PER_THREAD (F4_PER_BLOCK / HTHREADS) // 8

typedef __attribute__((ext_vector_type(2))) float v2f;
typedef __attribute__((ext_vector_type(8))) float v8f;

// ---------------------------------------------------------------- zero ws
__global__ void mi_zero(unsigned int* __restrict__ ws, int n) {
  int i = blockIdx.x * blockDim.x + threadIdx.x;
  if (i < n) ws[i] = 0u;
}

__device__ __forceinline__ int bin_of(float x) {
  // x in [-1,1] -> u=(x+1)/2 in [0,1]; bin = min(31, floor(u*32))
  int b = (int)((x + 1.0f) * 16.0f);
  return b > 31 ? 31 : b;
}

// ------------------------------------------------- streaming histogram
__global__ void mi_hist(const float* __restrict__ in,
                        const float* __restrict__ tg,
                        unsigned int* __restrict__ ghist) {
  __shared__ unsigned int lh[NBINS2];
  const int tid = threadIdx.x;
  for (int k = tid; k < NBINS2; k += HTHREADS) lh[k] = 0u;
  __syncthreads();

  const int s = blockIdx.x >> 6;   // sample index (64 chunks per sample)
  const int c = blockIdx.x & 63;   // chunk within sample

  const float4* u4 = (const float4*)in + (size_t)s * F4_PER + (size_t)c * F4_PER_BLOCK;
  const float4* v4 = (const float4*)tg + (size_t)s * F4_PER + (size_t)c * F4_PER_BLOCK;

  for (int j = 0; j < F4_PER_THREAD; ++j) {
    float4 x = u4[j * HTHREADS + tid];
    float4 y = v4[j * HTHREADS + tid];
#define DO_ELEM(ex, ey)                                            \
    do {                                                           \
      float xx = (ex), yy = (ey);                                  \
      if (fabsf(xx) <= 1.0f && fabsf(yy) <= 1.0f)                  \
        atomicAdd(&lh[bin_of(xx) * BINS + bin_of(yy)], 1u);        \
    } while (0)
    DO_ELEM(x.x, y.x); DO_ELEM(x.y, y.y);
    DO_ELEM(x.z, y.z); DO_ELEM(x.w, y.w);
#undef DO_ELEM
  }
  __syncthreads();

  unsigned int* gh = ghist + (size_t)s * NBINS2;
  for (int k = tid; k < NBINS2; k += HTHREADS) {
    unsigned int v = lh[k];
    if (v) atomicAdd(&gh[k], v);
  }
}

// ------------------------------------------- per-sample MI (1 wave32 each)
__global__ void mi_reduce(const unsigned int* __restrict__ ghist,
                          float* __restrict__ miArr) {
  __shared__ float lh[NBINS2];
  __shared__ float hx[BINS];   // row sums
  __shared__ float hy[BINS];   // col sums
  const int lane = threadIdx.x;          // 0..31, one full wave
  const int b = blockIdx.x;

  const unsigned int* gh = ghist + (size_t)b * NBINS2;
  for (int r = 0; r < BINS; ++r)
    lh[r * BINS + lane] = (float)gh[r * BINS + lane];
  __syncthreads();

  const int m  = lane & 15;              // M (or N) index within 16-wide tile
  const int kh = (lane >> 4) << 1;       // K sub-offset: 0 for lanes 0-15, 2 for 16-31

#if defined(__gfx1250__) && __has_builtin(__builtin_amdgcn_wmma_f32_16x16x4_f32)
  const v2f ones2 = {1.0f, 1.0f};
  // Row sums: Hx = H (32x32) * ones(32x16), accumulated over K in chunks of 4.
  for (int tile = 0; tile < 2; ++tile) {
    v8f acc = {};
    for (int ch = 0; ch < 8; ++ch) {
      v2f a;
      a.x = lh[(tile * 16 + m) * BINS + ch * 4 + kh];
      a.y = lh[(tile * 16 + m) * BINS + ch * 4 + kh + 1];
      acc = __builtin_amdgcn_wmma_f32_16x16x4_f32(false, a, false, ones2,
                                                  (short)0, acc, false, false);
    }
    // D vgpr r, lane l -> row (tile*16 + r + 8*(l>=16)), identical for every N.
    if (m == 0) {
      int base = tile * 16 + ((lane >> 4) << 3);
#pragma unroll
      for (int r = 0; r < 8; ++r) hx[base + r] = acc[r];
    }
  }
  // Col sums: Hy = ones(16x32) * H(32x32), per 16-wide column tile.
  for (int tile = 0; tile < 2; ++tile) {
    v8f acc = {};
    for (int ch = 0; ch < 8; ++ch) {
      v2f bv;
      bv.x = lh[(ch * 4 + kh) * BINS + tile * 16 + m];
      bv.y = lh[(ch * 4 + kh + 1) * BINS + tile * 16 + m];
      acc = __builtin_amdgcn_wmma_f32_16x16x4_f32(false, ones2, false, bv,
                                                  (short)0, acc, false, false);
    }
    // Every row of D equals the col-sum vector; lane l column = tile*16 + (l&15).
    if (lane < 16) hy[tile * 16 + m] = acc[0];
  }
#else
  if (lane < BINS) {
    float rs = 0.f, cs = 0.f;
    for (int k = 0; k < BINS; ++k) {
      rs += lh[lane * BINS + k];
      cs += lh[k * BINS + lane];
    }
    hx[lane] = rs;
    hy[lane] = cs;
  }
#endif
  __syncthreads();

  float T = 0.f;
  for (int i = 0; i < BINS; ++i) T += hx[i];
  const float logT = logf(T);

  // MI = (1/T) * sum_{h>0} h * (log h + log T - log Hx[i] - log Hy[j])
  float accmi = 0.f;
  const float hyl = hy[lane];
  const float loghy = (hyl > 0.f) ? logf(hyl) : 0.f;  // unused when column empty
  for (int i = 0; i < BINS; ++i) {
    float h = lh[i * BINS + lane];
    if (h > 0.f)
      accmi += h * (logf(h) + logT - logf(hx[i]) - loghy);
  }
  for (int off = 16; off > 0; off >>= 1)
    accmi += __shfl_xor(accmi, off, 32);
  if (lane == 0) miArr[b] = accmi / T;
}

// -------------------------------------- deterministic final mean, 1 wave
__global__ void mi_final(const float* __restrict__ miArr,
                         float* __restrict__ out) {
  const int lane = threadIdx.x;
  float v = miArr[lane];
  for (int off = 16; off > 0; off >>= 1)
    v += __shfl_xor(v, off, 32);
  if (lane == 0) out[0] = -(v / (float)BATCH);
}

// ----------------------------------------------------------------- launch
extern "C" void kernel_launch(void* const* d_in, const int* in_sizes, int n_in,
                              void* d_out, int out_size, void* d_ws, size_t ws_size,
                              hipStream_t stream) {
  (void)in_sizes; (void)n_in; (void)out_size; (void)ws_size;
  const float* in = (const float*)d_in[0];
  const float* tg = (const float*)d_in[1];

  unsigned int* ghist = (unsigned int*)d_ws;                       // 32*1024 u32
  float* miArr = (float*)((char*)d_ws + (size_t)BATCH * NBINS2 * sizeof(unsigned int));
  float* out = (float*)d_out;

  const int nz = BATCH * NBINS2 + BATCH;   // hist counts + mi array
  mi_zero<<<(nz + 255) / 256, 256, 0, stream>>>((unsigned int*)d_ws, nz);
  mi_hist<<<BATCH * CHUNKS, HTHREADS, 0, stream>>>(in, tg, ghist);
  mi_reduce<<<BATCH, 32, 0, stream>>>(ghist, miArr);
  mi_final<<<1, 32, 0, stream>>>(miArr, out);
}